// GAE_1546188226707
// MI455X (gfx1250) — compile-verified
//
#include <hip/hip_runtime.h>
#include <hip/hip_bf16.h>

// GAE backward scan, B=4096 rows x T=2048 steps.
// Strategy: 1 wave32 per row; lane i owns t = chunk*128 + 4*i .. +3 (float4,
// fully coalesced). Affine composition suffix-scan across the wave resolves
// the recurrence in parallel. Async global->LDS double buffering (CDNA5
// ASYNCcnt path) hides HBM latency; NT stores protect L2.

#define GAMMA  0.99f
#define LMBDA  0.95f
#define T_LEN  2048
#define WPB    8            // waves per 256-thread block
#define CHUNK  128          // t-values per wave iteration (32 lanes * 4)
#define NCHUNK (T_LEN / CHUNK)

typedef float v4f __attribute__((ext_vector_type(4)));
typedef int   v4i __attribute__((ext_vector_type(4)));

#if defined(__AMDGCN__) && __has_builtin(__builtin_amdgcn_global_load_async_to_lds_b128)
#define USE_ASYNC 1
// Builtin signature (from clang diagnostic): (global v4i*, shared v4i*, imm, imm)
typedef __attribute__((address_space(1))) v4i GV4;
typedef __attribute__((address_space(3))) v4i LV4;
#define ASYNC_B128(gptr, lptr) \
    __builtin_amdgcn_global_load_async_to_lds_b128((GV4*)(gptr), (LV4*)(lptr), 0, 0)
#else
#define USE_ASYNC 0
#endif

#if __has_builtin(__builtin_amdgcn_s_wait_asynccnt)
#define WAIT_ASYNCCNT(n) __builtin_amdgcn_s_wait_asynccnt(n)
#elif defined(__AMDGCN__)
#define WAIT_ASYNCCNT(n) asm volatile("s_wait_asynccnt " #n)
#else
#define WAIT_ASYNCCNT(n)
#endif

__global__ __launch_bounds__(256) void gae_scan_kernel(
    const float* __restrict__ reward,
    const int*   __restrict__ term,
    const float* __restrict__ value,
    const float* __restrict__ nextv,
    float* __restrict__ adv,
    float* __restrict__ ret,
    int B)
{
    const int lane = threadIdx.x & 31;
    const int wave = threadIdx.x >> 5;
    const int row  = blockIdx.x * WPB + wave;
    if (row >= B) return;

    const size_t base = (size_t)row * T_LEN;
    const float gl = GAMMA * LMBDA;

#if USE_ASYNC
    // [wave][buffer][array: r,t,v,nv][512 bytes] = 4 KB/wave, 32 KB/block
    __shared__ __align__(16) unsigned char smem[WPB][2][4][CHUNK * 4];

    // Prologue: stage chunk NCHUNK-1 into buffer 0 (one b128 per lane = 512B/array)
    {
        const size_t off = base + (size_t)(NCHUNK - 1) * CHUNK + lane * 4;
        ASYNC_B128(reward + off, &smem[wave][0][0][lane * 16]);
        ASYNC_B128(term + off,   &smem[wave][0][1][lane * 16]);
        ASYNC_B128(value + off,  &smem[wave][0][2][lane * 16]);
        ASYNC_B128(nextv + off,  &smem[wave][0][3][lane * 16]);
    }
#endif

    float carry = 0.0f;

    for (int c = NCHUNK - 1; c >= 0; --c) {
        v4f r4, v4, nv4;
        v4i t4;

#if USE_ASYNC
        const int p = (NCHUNK - 1 - c) & 1;
        if (c > 0) {
            // Issue next (earlier-in-time) chunk into the other buffer.
            const size_t off = base + (size_t)(c - 1) * CHUNK + lane * 4;
            ASYNC_B128(reward + off, &smem[wave][p ^ 1][0][lane * 16]);
            ASYNC_B128(term + off,   &smem[wave][p ^ 1][1][lane * 16]);
            ASYNC_B128(value + off,  &smem[wave][p ^ 1][2][lane * 16]);
            ASYNC_B128(nextv + off,  &smem[wave][p ^ 1][3][lane * 16]);
            WAIT_ASYNCCNT(4);   // async loads complete in order: current chunk done
        } else {
            WAIT_ASYNCCNT(0);
        }
        r4  = *(const v4f*)&smem[wave][p][0][lane * 16];
        t4  = *(const v4i*)&smem[wave][p][1][lane * 16];
        v4  = *(const v4f*)&smem[wave][p][2][lane * 16];
        nv4 = *(const v4f*)&smem[wave][p][3][lane * 16];
#else
        const size_t off = base + (size_t)c * CHUNK + lane * 4;
        r4  = __builtin_nontemporal_load((const v4f*)(reward + off));
        t4  = __builtin_nontemporal_load((const v4i*)(term + off));
        v4  = __builtin_nontemporal_load((const v4f*)(value + off));
        nv4 = __builtin_nontemporal_load((const v4f*)(nextv + off));
#endif

        // Per-element affine recurrence coefficients (time-ascending j=0..3):
        //   gae_t = a_t * gae_{t+1} + d_t
        float a[4], d[4];
#pragma unroll
        for (int j = 0; j < 4; ++j) {
            const float nd = 1.0f - (float)t4[j];
            a[j] = gl * nd;
            d[j] = r4[j] + GAMMA * nv4[j] * nd - v4[j];
        }

        // Thread-local composition f0 o f1 o f2 o f3 (applied to carry-from-later).
        float A = a[0] * a[1] * a[2] * a[3];
        float D = d[0] + a[0] * (d[1] + a[1] * (d[2] + a[2] * d[3]));

        // Wave32 inclusive *suffix* scan of affine composition:
        // after the loop, (A,D) at lane i == F_i o F_{i+1} o ... o F_31.
#pragma unroll
        for (int sh = 1; sh < 32; sh <<= 1) {
            const float Ah = __shfl_down(A, sh, 32);
            const float Dh = __shfl_down(D, sh, 32);
            if (lane + sh < 32) { D = D + A * Dh; A = A * Ah; }
        }

        // Exclusive input for this lane (carry flows from higher lanes).
        const float Ash = __shfl_down(A, 1, 32);
        const float Dsh = __shfl_down(D, 1, 32);
        float x = (lane == 31) ? carry : (Ash * carry + Dsh);

        // Apply local elements in reverse-time order; emit adv and ret.
        v4f av, rv;
#pragma unroll
        for (int j = 3; j >= 0; --j) {
            x = a[j] * x + d[j];
            av[j] = x;
            rv[j] = x + v4[j];
        }

        const size_t ooff = base + (size_t)c * CHUNK + lane * 4;
        __builtin_nontemporal_store(av, (v4f*)(adv + ooff));
        __builtin_nontemporal_store(rv, (v4f*)(ret + ooff));

        // New carry = gae at chunk start (lane 0's final x).
        carry = __shfl(x, 0, 32);
    }
}

extern "C" void kernel_launch(void* const* d_in, const int* in_sizes, int n_in,
                              void* d_out, int out_size, void* d_ws, size_t ws_size,
                              hipStream_t stream) {
    const float* reward = (const float*)d_in[0];
    const int*   term   = (const int*)d_in[1];
    const float* value  = (const float*)d_in[2];
    const float* nextv  = (const float*)d_in[3];

    const int BT = in_sizes[0];
    const int B  = BT / T_LEN;

    float* adv = (float*)d_out;            // output 0: advantages [B,T]
    float* ret = adv + (size_t)BT;         // output 1: returns    [B,T]

    dim3 block(256);
    dim3 grid((B + WPB - 1) / WPB);
    gae_scan_kernel<<<grid, block, 0, stream>>>(reward, term, value, nextv, adv, ret, B);
}